// FBPModule_11716670784236
// MI455X (gfx1250) — compile-verified
//
#include <hip/hip_runtime.h>
#include <hip/hip_bf16.h>

#define PI_F 3.14159265358979323846f
// Problem constants (match reference)
#define RESN 256
#define AANG 256
#define DDET 256
#define NSAM 512
#define CHALF 127.5f

typedef __attribute__((ext_vector_type(2))) float v2f;
typedef __attribute__((ext_vector_type(8))) float v8f;

// Types matching the async-to-LDS builtin signature exactly:
// param0: int __vector(4) addrspace(1)*   param1: int __vector(4) addrspace(3)*
typedef int v4i __attribute__((vector_size(16)));
typedef __attribute__((address_space(1))) v4i* gv4p;
typedef __attribute__((address_space(3))) v4i* lv4p;

#if __has_builtin(__builtin_amdgcn_global_load_async_to_lds_b128)
#define HAVE_ASYNC_COPY 1
#else
#define HAVE_ASYNC_COPY 0
#endif

// ---------------------------------------------------------------------------
// Bulk-stage a contiguous float buffer from global memory into LDS.
// On gfx1250 this uses GLOBAL_LOAD_ASYNC_TO_LDS_B128 (memory -> LDS direct,
// tracked by ASYNCcnt) instead of a VGPR round-trip. nfloats must be a
// multiple of 4 and 16B-aligned (true for all uses here).
// ---------------------------------------------------------------------------
__device__ __forceinline__ void lds_stage(float* __restrict__ lds,
                                          const float* __restrict__ g,
                                          int nfloats, int tid, int nthr) {
#if HAVE_ASYNC_COPY
    gv4p gp = (gv4p)g;                                 // addrspacecast + bitcast
    lv4p lp = (lv4p)lds;
    const int nvec = nfloats >> 2;                     // 16B chunks
    for (int i = tid; i < nvec; i += nthr) {
        __builtin_amdgcn_global_load_async_to_lds_b128(gp + i, lp + i, 0, 0);
    }
#if __has_builtin(__builtin_amdgcn_s_wait_asynccnt)
    __builtin_amdgcn_s_wait_asynccnt(0);
#else
    asm volatile("s_wait_asynccnt 0" ::: "memory");
#endif
#else
    for (int i = tid; i < nfloats; i += nthr) lds[i] = g[i];
#endif
}

// ---------------------------------------------------------------------------
// Kernel 1: build the TRANSPOSED circular-convolution ramp-filter matrix
//   filt[a,n] = sum_m sino[a,m] * H[m,n],  H[m,n] = (pi/256) * f[(n-m) mod 512]
//   f[0]=0.25 ; f[k]= -1/(pi*min(k,512-k))^2 for odd k ; else 0
// Stored as HT[n][m] so the GEMM's B operand reads are contiguous in K.
// ---------------------------------------------------------------------------
__global__ void build_h_kernel(float* __restrict__ HT) {
    int idx = blockIdx.x * blockDim.x + threadIdx.x;   // 65536 entries
    int n = idx >> 8;                                  // output detector bin
    int m = idx & 255;                                 // input detector bin (K)
    int k = (n - m) & 511;
    float v;
    if (k == 0) {
        v = 0.25f;
    } else if (k & 1) {
        int kk = (k < 256) ? k : (512 - k);
        float pk = PI_F * (float)kk;
        v = -1.0f / (pk * pk);
    } else {
        v = 0.0f;
    }
    HT[idx] = v * (PI_F / 256.0f);   // 2*f * (pi/(2*A)) folded together
}

// ---------------------------------------------------------------------------
// Kernel 2: fan-beam forward projection with subsample mask folded in.
// One block = 512 threads = 2 angles x 256 detectors. Image staged in LDS.
// ---------------------------------------------------------------------------
__device__ __forceinline__ float tap2d(const float* __restrict__ s, int iy, int ix) {
    bool in = ((unsigned)ix < 256u) && ((unsigned)iy < 256u);
    int cy = min(max(iy, 0), 255);
    int cx = min(max(ix, 0), 255);
    float v = s[cy * 256 + cx];
    return in ? v : 0.0f;
}

__global__ __launch_bounds__(512) void fp_kernel(const float* __restrict__ img,
                                                 const int* __restrict__ ssf_p,
                                                 float* __restrict__ sparse) {
    extern __shared__ float simg[];                    // 256*256 floats = 256KB
    const int tid = threadIdx.x;                       // 0..511
    __builtin_prefetch(img + tid, 0, 0);               // global_prefetch_b8
    lds_stage(simg, img, RESN * RESN, tid, 512);
    __syncthreads();

    const int a = blockIdx.x * 2 + (tid >> 8);         // angle
    const int d = tid & 255;                           // detector
    const int ssf = *ssf_p;
    if ((a % ssf) != 0) {                              // masked angle: row is zero
        sparse[a * DDET + d] = 0.0f;
        return;
    }

    float beta = (float)a * (2.0f * PI_F / (float)AANG);
    float sb, cb;
    __sincosf(beta, &sb, &cb);
    float u  = ((float)d - 127.5f) * 2.0f;             // SP = 2
    float sx = -256.0f * sb, sy = 256.0f * cb;         // SD = 256
    float dx = u * cb + 256.0f * sb;                   // DD = 256
    float dy = u * sb - 256.0f * cb;
    float rx = dx - sx, ry = dy - sy;
    float bx = sx + CHALF, by = sy + CHALF;

    float acc = 0.0f;
#pragma unroll 4
    for (int i = 0; i < NSAM; ++i) {
        float t = ((float)i + 0.5f) * (1.0f / (float)NSAM);
        float x = bx + t * rx;
        float y = by + t * ry;
        float xf = floorf(x), yf = floorf(y);
        float fx = x - xf, fy = y - yf;
        int ix = (int)xf, iy = (int)yf;
        float v00 = tap2d(simg, iy,     ix);
        float v01 = tap2d(simg, iy,     ix + 1);
        float v10 = tap2d(simg, iy + 1, ix);
        float v11 = tap2d(simg, iy + 1, ix + 1);
        float top = v00 + fx * (v01 - v00);
        float bot = v10 + fx * (v11 - v10);
        acc += top + fy * (bot - top);
    }
    float step = sqrtf(u * u + 512.0f * 512.0f) * (1.0f / (float)NSAM);  // SDD=512
    sparse[a * DDET + d] = acc * step;
}

// ---------------------------------------------------------------------------
// Kernel 3: ramp filter as GEMM  F[256x256] = S[256x256] * HT[256x256]^T
// via V_WMMA_F32_16X16X4_F32. One wave32 per 16x16 tile, K-loop of 64 WMMAs.
// A layout (16x4 f32): lanes 0-15 hold M=0..15 with K={0,1} in VGPR{0,1};
//                      lanes 16-31 hold K={2,3}. B mirrors with N across lanes.
// Both operands are contiguous K-pairs -> single global_load_b64 each.
// C/D: VGPR r -> (M=r, N=lane) for lanes 0-15, (M=r+8) for lanes 16-31.
// ---------------------------------------------------------------------------
__global__ __launch_bounds__(256) void gemm_filter_kernel(const float* __restrict__ S,
                                                          const float* __restrict__ HT,
                                                          float* __restrict__ F) {
    const int lane = threadIdx.x & 31;
    const int wave = threadIdx.x >> 5;                 // 0..7
    const int tile = blockIdx.x * 8 + wave;            // 0..255 tiles
    const int tm = (tile >> 4) * 16;                   // tile row (angles)
    const int tn = (tile & 15) * 16;                   // tile col (detector out)
    const int r    = lane & 15;                        // M (or N) within tile
    const int half = lane >> 4;                        // K-pair selector

    const float* arow = S  + (tm + r) * 256;           // A row (angle tm+r)
    const float* brow = HT + (tn + r) * 256;           // B col (detector tn+r)

    v8f c = {};
    for (int k0 = 0; k0 < 256; k0 += 4) {
        const int kb = k0 + half * 2;                  // even -> 8B aligned
        v2f av = *(const v2f*)(arow + kb);
        v2f bv = *(const v2f*)(brow + kb);
        c = __builtin_amdgcn_wmma_f32_16x16x4_f32(
                /*neg_a=*/false, av, /*neg_b=*/false, bv,
                /*c_mod=*/(short)0, c, /*reuse_a=*/false, /*reuse_b=*/false);
    }
#pragma unroll
    for (int rr = 0; rr < 8; ++rr) {
        F[(tm + rr + half * 8) * 256 + tn + r] = c[rr];
    }
}

// ---------------------------------------------------------------------------
// Kernel 4: weighted fan-beam backprojection. Filtered sinogram staged in LDS
// (+ small trig table). One thread per output pixel; skip masked angles
// (their filtered rows are exactly zero).
// ---------------------------------------------------------------------------
__global__ __launch_bounds__(512) void bp_kernel(const float* __restrict__ F,
                                                 const int* __restrict__ ssf_p,
                                                 float* __restrict__ out) {
    extern __shared__ float sf[];                      // 65536 filt + 512 trig
    float* strig = sf + RESN * RESN;                   // cb[256], sb[256]
    const int tid = threadIdx.x;
    __builtin_prefetch(F + tid, 0, 0);                 // global_prefetch_b8
    lds_stage(sf, F, RESN * RESN, tid, 512);
    if (tid < 256) {
        float beta = (float)tid * (2.0f * PI_F / (float)AANG);
        float s, cc;
        __sincosf(beta, &s, &cc);
        strig[tid] = cc;
        strig[256 + tid] = s;
    }
    __syncthreads();

    const int p  = blockIdx.x * 512 + tid;             // pixel id
    const int px = p & 255;
    const int py = p >> 8;
    float x = (float)px - CHALF;
    float y = (float)py - CHALF;
    const int ssf = *ssf_p;

    float acc = 0.0f;
    for (int a = 0; a < AANG; a += ssf) {
        float cb = strig[a], sb = strig[256 + a];
        float pu =  x * cb + y * sb;
        float pv = -x * sb + y * cb;
        float inv = 1.0f / (256.0f - pv);              // SD - pv
        float s = pu * 256.0f * inv + 127.5f;          // pu*SDD/((SD-pv)*SP)
        float w = 256.0f * inv;                        // SD/(SD-pv)
        w = w * w;
        float s0 = floorf(s);
        float fr = s - s0;
        int i0 = (int)s0;
        int i1 = i0 + 1;
        float g0 = ((unsigned)i0 < 256u) ? sf[a * 256 + min(max(i0, 0), 255)] : 0.0f;
        float g1 = ((unsigned)i1 < 256u) ? sf[a * 256 + min(max(i1, 0), 255)] : 0.0f;
        acc += w * (g0 + fr * (g1 - g0));
    }
    out[p] = acc * (2.0f * PI_F / (float)AANG);
}

// ---------------------------------------------------------------------------
extern "C" void kernel_launch(void* const* d_in, const int* in_sizes, int n_in,
                              void* d_out, int out_size, void* d_ws, size_t ws_size,
                              hipStream_t stream) {
    const float* img = (const float*)d_in[0];          // [1,1,256,256] f32
    const int*   ssf = (const int*)d_in[1];            // scalar int
    float* out = (float*)d_out;                        // [1,1,256,256] f32

    float* sparse = (float*)d_ws;                      // 65536 f32: masked sinogram
    float* HTmat  = sparse + RESN * RESN;              // 65536 f32: filter matrix (transposed)
    float* filt   = HTmat  + RESN * RESN;              // 65536 f32: filtered sinogram

    // 1) filter matrix (input-independent, rebuilt each call: stateless)
    build_h_kernel<<<256, 256, 0, stream>>>(HTmat);
    // 2) forward projection with subsample mask; image staged in 256KB LDS
    fp_kernel<<<128, 512, RESN * RESN * sizeof(float), stream>>>(img, ssf, sparse);
    // 3) ramp filter as f32 WMMA GEMM
    gemm_filter_kernel<<<32, 256, 0, stream>>>(sparse, HTmat, filt);
    // 4) backprojection; filtered sinogram staged in LDS
    bp_kernel<<<128, 512, (RESN * RESN + 512) * sizeof(float), stream>>>(filt, ssf, out);
}